// VectorQuantizer_24223615550166
// MI455X (gfx1250) — compile-verified
//
#include <hip/hip_runtime.h>
#include <cstdint>
#include <cstddef>

// ---------------- problem constants ----------------
constexpr int    KC   = 24;        // codebook entries
constexpr int    DIM  = 128;       // embedding dim
constexpr float  BETA  = 0.5f;
constexpr float  DECAY = 0.95f;
constexpr int    NB   = 524288;    // batch

typedef __attribute__((ext_vector_type(2))) float v2f;
typedef __attribute__((ext_vector_type(8))) float v8f;

// ---------------- launch geometry ----------------
constexpr int WAVES          = 4;               // 128 threads / block
constexpr int BLOCK          = WAVES * 32;
constexpr int ROWS_PER_BLOCK = 512;             // 32 tiles of 16 rows
constexpr int TILES_PER_BLK  = ROWS_PER_BLOCK / 16;
constexpr int NBLOCKS        = NB / ROWS_PER_BLOCK;   // 1024

// ---------------- dynamic LDS layout (floats) ----------------
constexpr int CB_STRIDE = 132;   // padded row stride (528B: 16B aligned, bank-staggered)
constexpr int Z_STRIDE  = 132;

constexpr int CB_OFF   = 0;                              // 32 rows (24..31 zeroed)
constexpr int WN2_OFF  = CB_OFF + 32 * CB_STRIDE;        // 32
constexpr int NCS_OFF  = WN2_OFF + 32;                   // 32
constexpr int LOSS_OFF = NCS_OFF + 32;                   // 4
constexpr int NES_OFF  = LOSS_OFF + 4;                   // KC*DIM
constexpr int ZT_OFF   = NES_OFF + KC * DIM;             // WAVES*16*Z_STRIDE
constexpr int S_OFF    = ZT_OFF + WAVES * 16 * Z_STRIDE; // WAVES*16*32
constexpr int WT_OFF   = S_OFF + WAVES * 16 * 32;        // WAVES*16*24
constexpr int IDX_OFF  = WT_OFF + WAVES * 16 * 24;       // WAVES*16 (ints)
constexpr int SMEM_FLOATS = IDX_OFF + WAVES * 16;
constexpr size_t SMEM_BYTES = (size_t)SMEM_FLOATS * 4;   // ~76 KB (<320 KB WGP LDS)

// ---------------- output layout (concatenated, float32) ----------------
constexpr size_t OFF_SOFT = 0;
constexpr size_t OFF_HARD = (size_t)NB * DIM;
constexpr size_t OFF_IDX  = 2ull * (size_t)NB * DIM;
constexpr size_t OFF_W    = OFF_IDX + (size_t)NB;
constexpr size_t OFF_LOSS = OFF_W + (size_t)NB * KC;
constexpr size_t OFF_CB   = OFF_LOSS + 1;
constexpr size_t OFF_ECS  = OFF_CB + (size_t)KC * DIM;
constexpr size_t OFF_EES  = OFF_ECS + (size_t)KC;

// ---------------- workspace layout (floats) ----------------
constexpr int WS_NCS    = 0;                 // 32
constexpr int WS_NES    = 32;                // KC*DIM
constexpr int WS_LOSS   = WS_NES + KC * DIM; // 1
constexpr int WS_FLOATS = WS_LOSS + 1;

__global__ void vq_zero(float* __restrict__ ws) {
  const int i = blockIdx.x * blockDim.x + threadIdx.x;
  if (i < WS_FLOATS) ws[i] = 0.0f;
}

__global__ __launch_bounds__(BLOCK)
void vq_main(const float* __restrict__ z_e,
             const float* __restrict__ codebook,
             float* __restrict__ out,
             float* __restrict__ ws) {
  extern __shared__ float smem[];
  float* cb    = smem + CB_OFF;
  float* wn2   = smem + WN2_OFF;
  float* ncs   = smem + NCS_OFF;
  float* lossl = smem + LOSS_OFF;
  float* nes   = smem + NES_OFF;
  int*   idxs  = (int*)(smem + IDX_OFF);

  const int tid  = threadIdx.x;
  const int wave = tid >> 5;
  const int lane = tid & 31;
  const int r    = lane & 15;      // row (A) or column (B/C) index
  const int hi   = lane >> 4;      // half-wave select
  const int ko   = hi * 2;         // K-pair offset within a 4-wide chunk

  float* Zl  = smem + ZT_OFF + wave * 16 * Z_STRIDE;
  float* Sl  = smem + S_OFF  + wave * 16 * 32;
  float* Wtl = smem + WT_OFF + wave * 16 * 24;
  int*   idl = idxs + wave * 16;

  // ---- block init: zero accumulators + padded codebook, code norms ----
  for (int i = tid; i < 32 * CB_STRIDE; i += BLOCK) cb[i] = 0.0f;
  for (int i = tid; i < KC * DIM; i += BLOCK) nes[i] = 0.0f;
  if (tid < 32) ncs[tid] = 0.0f;
  if (tid == 0) lossl[0] = 0.0f;
  __syncthreads();
  for (int i = tid; i < KC * DIM; i += BLOCK)
    cb[(i >> 7) * CB_STRIDE + (i & 127)] = codebook[i];
  if (tid < KC) {
    float s = 0.0f;
    for (int d = 0; d < DIM; ++d) { const float v = codebook[tid * DIM + d]; s += v * v; }
    wn2[tid] = s;
  }
  __syncthreads();

  // ---- per-wave 16-row tiles (all LDS deps wave-internal -> no barriers) ----
  for (int t = wave; t < TILES_PER_BLK; t += WAVES) {
    const size_t row0 = (size_t)blockIdx.x * ROWS_PER_BLOCK + (size_t)t * 16;

    // ---- stage z_e tile 16x128 with async global->LDS DMA (ASYNCcnt path) ----
    // WAR guard: previous tile's ds_loads of Zl must retire before async writes land
    asm volatile("s_wait_dscnt 0x0" ::: "memory");
    {
      // dynamic LDS starts at offset 0 (no static __shared__ in this kernel)
      const unsigned lds0 = (unsigned)((ZT_OFF + wave * 16 * Z_STRIDE) * 4)
                          + (unsigned)(lane * 16);
      #pragma unroll
      for (int rr = 0; rr < 16; ++rr) {
        const float* gp = &z_e[(row0 + rr) * DIM + lane * 4];
        const unsigned la = lds0 + (unsigned)(rr * Z_STRIDE * 4);
        asm volatile("global_load_async_to_lds_b128 %0, %1, off"
                     :: "v"(la), "v"(gp) : "memory");
      }
    }
    asm volatile("s_wait_asynccnt 0x0" ::: "memory");

    // GEMM1: S[16x32] = Z[16x128] * codebook^T ; 64x v_wmma_f32_16x16x4_f32
    v8f c0 = {};
    v8f c1 = {};
    {
      const float* za  = &Zl[r * Z_STRIDE + ko];
      const float* b0p = &cb[r * CB_STRIDE + ko];           // codes 0..15
      const float* b1p = &cb[(16 + r) * CB_STRIDE + ko];    // codes 16..31 (24..31 == 0)
      #pragma unroll
      for (int k4 = 0; k4 < DIM; k4 += 4) {
        v2f a;  a.x  = za[k4];  a.y  = za[k4 + 1];
        v2f b0; b0.x = b0p[k4]; b0.y = b0p[k4 + 1];
        v2f b1; b1.x = b1p[k4]; b1.y = b1p[k4 + 1];
        c0 = __builtin_amdgcn_wmma_f32_16x16x4_f32(false, a, false, b0, (short)0, c0, false, false);
        c1 = __builtin_amdgcn_wmma_f32_16x16x4_f32(false, a, false, b1, (short)0, c1, false, false);
      }
    }
    #pragma unroll
    for (int v = 0; v < 8; ++v) {
      Sl[(v + 8 * hi) * 32 + r]      = c0[v];
      Sl[(v + 8 * hi) * 32 + 16 + r] = c1[v];
    }

    // per-row: argmin + softmax (||z||^2 cancels; TEMP == 1)
    if (lane < 16) {
      const float* Sr = &Sl[lane * 32];
      float dv[KC];
      float dmin = 3.4e38f;
      int   imin = 0;
      #pragma unroll
      for (int n = 0; n < KC; ++n) {
        const float v = wn2[n] - 2.0f * Sr[n];
        dv[n] = v;
        if (v < dmin) { dmin = v; imin = n; }   // strict '<' == first-min (jnp.argmin)
      }
      float ssum = 0.0f;
      #pragma unroll
      for (int n = 0; n < KC; ++n) { const float e = __expf(dmin - dv[n]); dv[n] = e; ssum += e; }
      const float inv = 1.0f / ssum;
      float* Wr = &Wtl[lane * 24];
      float* gw = &out[OFF_W + (row0 + lane) * KC];
      #pragma unroll
      for (int n = 0; n < KC; ++n) { const float w = dv[n] * inv; Wr[n] = w; gw[n] = w; }
      idl[lane] = imin;
      out[OFF_IDX + row0 + lane] = (float)imin;
      atomicAdd(&ncs[imin], 1.0f);
    }

    // GEMM2: z_q_soft[16x128] = W[16x24] * codebook[24x128] ; 48x v_wmma
    {
      const float* wa = &Wtl[r * 24 + ko];
      #pragma unroll
      for (int nt = 0; nt < 8; ++nt) {
        v8f cc = {};
        const int d0 = nt * 16 + r;
        #pragma unroll
        for (int k4 = 0; k4 < KC; k4 += 4) {
          v2f a; a.x = wa[k4]; a.y = wa[k4 + 1];
          v2f b; b.x = cb[(k4 + ko) * CB_STRIDE + d0];
                 b.y = cb[(k4 + ko + 1) * CB_STRIDE + d0];
          cc = __builtin_amdgcn_wmma_f32_16x16x4_f32(false, a, false, b, (short)0, cc, false, false);
        }
        #pragma unroll
        for (int v = 0; v < 8; ++v)
          out[OFF_SOFT + (row0 + (size_t)(v + 8 * hi)) * DIM + d0] = cc[v];
      }
    }

    // z_q_hard gather + segment-sum (LDS atomics) + elementwise loss
    float lacc = 0.0f;
    for (int rr = 0; rr < 16; ++rr) {
      const int kq = idl[rr];
      const float4 cv = *(const float4*)&cb[kq * CB_STRIDE + lane * 4];
      *(float4*)&out[OFF_HARD + (row0 + rr) * DIM + lane * 4] = cv;
      const float4 zv = *(const float4*)&Zl[rr * Z_STRIDE + lane * 4];
      const float dx = cv.x - zv.x, dy = cv.y - zv.y;
      const float dz = cv.z - zv.z, dw = cv.w - zv.w;
      lacc += dx * dx + dy * dy + dz * dz + dw * dw;
      float* np = &nes[kq * DIM + lane * 4];
      atomicAdd(&np[0], zv.x);
      atomicAdd(&np[1], zv.y);
      atomicAdd(&np[2], zv.z);
      atomicAdd(&np[3], zv.w);
    }
    atomicAdd(lossl, lacc);
  }

  // ---- flush block-local accumulators (few global atomics per block) ----
  __syncthreads();
  if (tid < KC) atomicAdd(&ws[WS_NCS + tid], ncs[tid]);
  for (int i = tid; i < KC * DIM; i += BLOCK) atomicAdd(&ws[WS_NES + i], nes[i]);
  if (tid == 0) atomicAdd(&ws[WS_LOSS], lossl[0]);
}

__global__ void vq_finalize(const float* __restrict__ z_e,
                            const float* __restrict__ ema_cs,
                            const float* __restrict__ ema_es,
                            const int* __restrict__ rand_idx,
                            const float* __restrict__ ws,
                            float* __restrict__ out) {
  __shared__ float ecs_s[KC];
  __shared__ float smoothed[KC];
  __shared__ float nsum;
  const int tid = threadIdx.x;
  if (tid < KC) ecs_s[tid] = DECAY * ema_cs[tid] + (1.0f - DECAY) * ws[WS_NCS + tid];
  __syncthreads();
  if (tid == 0) {
    float n = 0.0f;
    for (int k = 0; k < KC; ++k) n += ecs_s[k];
    nsum = n;
  }
  __syncthreads();
  if (tid < KC) smoothed[tid] = (ecs_s[tid] + 1e-5f) / (nsum + KC * 1e-5f) * nsum;
  __syncthreads();
  for (int i = tid; i < KC * DIM; i += blockDim.x) {
    const int k = i >> 7, d = i & 127;
    const float ees_v = DECAY * ema_es[i] + (1.0f - DECAY) * ws[WS_NES + i];
    const bool dead = ecs_s[k] < 0.1f;
    const float rs = z_e[(size_t)rand_idx[k] * DIM + d];
    out[OFF_CB + i]  = dead ? rs : ees_v / smoothed[k];
    out[OFF_EES + i] = dead ? rs : ees_v;
  }
  if (tid < KC) out[OFF_ECS + tid] = (ecs_s[tid] < 0.1f) ? 1.0f : ecs_s[tid];
  if (tid == 0)
    out[OFF_LOSS] = (1.0f + BETA) * ws[WS_LOSS] / ((float)NB * (float)DIM);
}

extern "C" void kernel_launch(void* const* d_in, const int* in_sizes, int n_in,
                              void* d_out, int out_size, void* d_ws, size_t ws_size,
                              hipStream_t stream) {
  (void)in_sizes; (void)n_in; (void)out_size; (void)ws_size;
  const float* z_e      = (const float*)d_in[0];
  const float* codebook = (const float*)d_in[1];
  const float* ema_cs   = (const float*)d_in[2];
  const float* ema_es   = (const float*)d_in[3];
  const int*   rand_idx = (const int*)d_in[4];
  float* out = (float*)d_out;
  float* ws  = (float*)d_ws;

  vq_zero<<<(WS_FLOATS + 255) / 256, 256, 0, stream>>>(ws);
  vq_main<<<NBLOCKS, BLOCK, SMEM_BYTES, stream>>>(z_e, codebook, out, ws);
  vq_finalize<<<1, 256, 0, stream>>>(z_e, ema_cs, ema_es, rand_idx, ws, out);
}